// NonAdaptiveLearningMask_51848845197804
// MI455X (gfx1250) — compile-verified
//
#include <hip/hip_runtime.h>
#include <cstdint>

typedef float f4 __attribute__((ext_vector_type(4)));

#define B_ 16
#define F_ 256
#define T_ 8192
#define K_ 204   // int(256 * (1 - 0.2))

// ---------------------------------------------------------------------------
// Kernel A: binary[b,f] = (rank of x2[b,f] within row b is < K) && (x2 > 0)
// x2 = sigmoid(w[f]) + 0.05*noise[b,f].  16 blocks x 256 threads; row in LDS.
// rank-by-count is exactly equivalent to (x2 >= kth-largest), ties included.
// ---------------------------------------------------------------------------
__global__ void compute_binary_kernel(const float* __restrict__ w_mask,
                                      const float* __restrict__ noise,
                                      float* __restrict__ bin) {
    __shared__ float sh[F_];
    const int b = blockIdx.x;
    const int f = threadIdx.x;
    const float w   = w_mask[f];
    const float sig = 1.0f / (1.0f + expf(-w));
    const float x2  = sig + noise[b * F_ + f] * 0.05f;
    sh[f] = x2;
    __syncthreads();
    int cnt = 0;
#pragma unroll 8
    for (int j = 0; j < F_; ++j)
        cnt += (sh[j] > x2) ? 1 : 0;          // LDS broadcast reads, conflict-free
    bin[b * F_ + f] = (cnt < K_ && x2 > 0.0f) ? 1.0f : 0.0f;
}

// ---------------------------------------------------------------------------
// Kernel B: out1[(b*F+f)*T + t] = bin[b*F+f] * x[(b*F+f)*T + t]
// Streaming scale-copy. 1024 float4 per block (4096 floats) => a block never
// crosses a (b,f) row (row = 2048 float4), so the mask value is block-uniform
// and lives in an SGPR. NT loads + NT stores; prefetch ahead.
// ---------------------------------------------------------------------------
__global__ void masked_x_kernel(const float* __restrict__ x,
                                const float* __restrict__ bin,
                                float* __restrict__ out) {
    const long long base4 = (long long)blockIdx.x * 1024;   // float4 index
    const int bf = (int)(base4 >> 11);                      // uniform: /2048
    const float s = bin[bf];                                // scalar (SGPR) load
    const f4* __restrict__ xin = (const f4*)x;
    f4* __restrict__ o = (f4*)out;
    const long long i0 = base4 + threadIdx.x;

    // speculative stream-ahead prefetch (two blocks ahead)
    const long long pf = i0 + 2048;
    if (pf < (long long)B_ * F_ * (T_ / 4))
        __builtin_prefetch((const void*)&xin[pf], 0, 0);

#pragma unroll
    for (int k = 0; k < 4; ++k) {
        const long long i = i0 + (long long)k * 256;
        f4 v = __builtin_nontemporal_load(&xin[i]);
        f4 r = v * s;
        __builtin_nontemporal_store(r, &o[i]);
    }
}

// ---------------------------------------------------------------------------
// Kernel C: out2[(b*T + t)*F + f] = bin[b*F + f]   (tile 1KiB row 8192x)
// Stage the 1 KiB row into LDS with CDNA5 async load-to-LDS (ASYNCcnt path),
// then each thread replays its float4 with NT b128 stores.
// Grid: (128 chunks of 64 t-rows) x (16 batches); 256 threads.
// ---------------------------------------------------------------------------
__global__ void binary_mask_kernel(const float* __restrict__ bin,
                                   float* __restrict__ out) {
    __shared__ __align__(16) float sh[F_];
    const int b     = blockIdx.y;
    const int chunk = blockIdx.x;
    const int tid   = threadIdx.x;

    // 64 lanes x b128 = 1 KiB: global -> LDS, no VGPR round-trip
    if (tid < 64) {
        uint32_t lds_off = (uint32_t)(uint64_t)(&sh[tid * 4]);
        uint64_t gaddr   = (uint64_t)(const void*)(bin + b * F_ + tid * 4);
        asm volatile("global_load_async_to_lds_b128 %0, %1, off"
                     :: "v"(lds_off), "v"(gaddr) : "memory");
    }
    asm volatile("s_wait_asynccnt 0" ::: "memory");
    __syncthreads();

    const f4 v = ((const f4*)sh)[tid & 63];   // one float4 of the mask row

    // this block writes t in [chunk*64, chunk*64+64)
    const int trow = tid >> 6;                // 0..3
    const int fo   = tid & 63;                // float4 index along F
    f4* __restrict__ o = (f4*)out;
    const long long o4base = (long long)b * (T_ * (F_ / 4))
                           + (long long)chunk * 64 * (F_ / 4)
                           + (long long)trow * (F_ / 4) + fo;
#pragma unroll
    for (int i = 0; i < 16; ++i)              // t advances by 4 per iter
        __builtin_nontemporal_store(v, &o[o4base + (long long)i * 4 * (F_ / 4)]);
}

// ---------------------------------------------------------------------------
extern "C" void kernel_launch(void* const* d_in, const int* in_sizes, int n_in,
                              void* d_out, int out_size, void* d_ws, size_t ws_size,
                              hipStream_t stream) {
    const float* x     = (const float*)d_in[0];   // (16,1,256,8192) f32
    const float* wmask = (const float*)d_in[1];   // (1,1,1,256)    f32
    const float* noise = (const float*)d_in[2];   // (16,1,1,256)   f32

    float* out1 = (float*)d_out;                            // masked_x, 33.5M f32
    float* out2 = out1 + (long long)B_ * F_ * T_;           // binary_mask, 33.5M f32
    float* bin  = (float*)d_ws;                             // 16 KiB scratch

    compute_binary_kernel<<<B_, F_, 0, stream>>>(wmask, noise, bin);

    // 33,554,432 floats / (1024 f4/block * 4) = 8192 blocks
    masked_x_kernel<<<8192, 256, 0, stream>>>(x, bin, out1);

    binary_mask_kernel<<<dim3(T_ / 64, B_), 256, 0, stream>>>(bin, out2);
}